// Walker_12601434046503
// MI455X (gfx1250) — compile-verified
//
#include <hip/hip_runtime.h>
#include <hip/hip_bf16.h>

// Problem constants (match reference file)
#define N_NODES 50000
#define DEG     32
#define STEPS   64
#define WIN     8
#define WORDS   794   // (50000 + 62) / 63

// One walk step. LIM = number of leading invalid window rows (compile-time
// constant: WIN-(i+1) during ramp-up, 0 in steady state). FIRST disables the
// backtrack test (reference: bt &= i > 0).
//
// All index-carrying inputs hold values < 2^31, so we read only the low dword
// of each int64 (little-endian) -> 32-bit gathers and 32-bit chain arithmetic.
// choices (values < 2^62) and adj_bits (full bitsets) stay 64-bit.
template <int LIM, bool FIRST>
static __device__ __forceinline__ void walk_step(
    int i, int v, int* window,
    const unsigned* __restrict__ adj_nodes32,   // low dwords of edge_index[1]
    const unsigned* __restrict__ adj_offset32,
    const unsigned* __restrict__ degrees32,
    const unsigned* __restrict__ node_id32,
    const long long* __restrict__ adj_bits,
    const long long* __restrict__ choices,
    float* __restrict__ xrow, float* __restrict__ nrow,
    float* __restrict__ erow)
{
    const unsigned long long ch = (unsigned long long)
        __builtin_nontemporal_load(&choices[(size_t)i * N_NODES + v]);
    // Branch-free prefetch of the next (coalesced) choices row.
    const int ip = (i + 1 < STEPS) ? (i + 1) : i;
    __builtin_prefetch(&choices[(size_t)ip * N_NODES + v], 0, 0);

    const int cur = window[WIN - 1];
    const int old = window[WIN - 2];

    const unsigned deg = degrees32[2 * (size_t)cur];
    const unsigned off = adj_offset32[2 * (size_t)cur];
    const unsigned nbd = (deg == 1u) ? deg : (deg - 1u);

    // Hot modulo: power-of-two degrees (true for this dataset) -> single AND,
    // exact in 32 bits because mask < 2^32. Generic 64-bit urem stays for
    // correctness but is EXEC-skipped when unused.
    const bool pow2 = (deg & (deg - 1u)) == 0u;
    const unsigned eidx = pow2 ? ((unsigned)ch & (deg - 1u))
                               : (unsigned)(ch % (unsigned long long)deg);

    unsigned chosen = off + eidx;
    int nw = (int)adj_nodes32[2 * (size_t)chosen];

    // Backtrack avoidance, exactly as the reference. Rare (prob ~DEG/N).
    if (!FIRST && nw == old) {
        // eidx + 1 + (ch % nbd) <= 2*deg - 2 < 2*deg: one conditional
        // subtract is exactly the final "% deg".
        unsigned ai = eidx + 1u + (unsigned)(ch % (unsigned long long)nbd);
        if (ai >= deg) ai -= deg;
        chosen = off + ai;
        nw = (int)adj_nodes32[2 * (size_t)chosen];
    }

    // gfx1250 prefetch path: next step's dependent loads are the only
    // address-predictable latency on the serial chain.
    __builtin_prefetch(&degrees32[2 * (size_t)nw], 0, 1);
    __builtin_prefetch(&adj_offset32[2 * (size_t)nw], 0, 1);
    __builtin_prefetch(&adj_nodes32[2 * ((size_t)nw * DEG)], 0, 1);       // +0B
    __builtin_prefetch(&adj_nodes32[2 * ((size_t)nw * DEG + 16)], 0, 1);  // +128B

    const unsigned cid  = node_id32[2 * (size_t)nw];
    const unsigned word = cid / 63u;   // 32-bit magic multiply
    const unsigned bit  = cid % 63u;

#pragma unroll
    for (int j = 0; j < WIN; ++j) {
        const bool valid = (j >= LIM);        // compile-time during ramp-up
        const bool idr = valid && (window[j] == nw);
        __builtin_nontemporal_store(idr ? 1.0f : 0.0f,
                                    &xrow[j * 65 + (i + 1)]);
        if (j < WIN - 1) {                    // reference slices off row 7
            bool adr = false;
            if (valid) {                      // steady state: unconditional
                const unsigned idx =
                    (unsigned)window[j] * (unsigned)WORDS + word;
                const long long val = adj_bits[idx];
                adr = (((val >> bit) & 1LL) == 1LL);
            }
            __builtin_nontemporal_store(adr ? 1.0f : 0.0f,
                                        &xrow[(WIN + j) * 65 + (i + 1)]);
        }
    }

#pragma unroll
    for (int j = 0; j < WIN - 1; ++j) window[j] = window[j + 1];
    window[WIN - 1] = nw;

    __builtin_nontemporal_store((float)nw,     &nrow[i + 1]);
    __builtin_nontemporal_store((float)chosen, &erow[i]);
}

__global__ __launch_bounds__(256) void Walker_12601434046503_kernel(
    const unsigned* __restrict__ adj_nodes32,
    const unsigned* __restrict__ adj_offset32,
    const unsigned* __restrict__ degrees32,
    const unsigned* __restrict__ node_id32,
    const long long* __restrict__ adj_bits,
    const long long* __restrict__ choices,
    float* __restrict__ out_nodes,             // (N, 65)
    float* __restrict__ out_edges,             // (N, 64)
    float* __restrict__ out_x)                 // (N, 15, 65)
{
    const int v = blockIdx.x * blockDim.x + threadIdx.x;
    if (v >= N_NODES) return;

    int window[WIN];
#pragma unroll
    for (int j = 0; j < WIN; ++j) window[j] = -1;
    window[WIN - 1] = v;

    float* __restrict__ xrow = out_x + (size_t)v * 15 * 65;
    float* __restrict__ nrow = out_nodes + (size_t)v * 65;
    float* __restrict__ erow = out_edges + (size_t)v * 64;

    // t == 0 columns: walk starts at v; feature columns are zero.
    __builtin_nontemporal_store((float)v, &nrow[0]);
#pragma unroll
    for (int k = 0; k < 15; ++k)
        __builtin_nontemporal_store(0.0f, &xrow[k * 65]);

    // Ramp-up: LIM = WIN - (i+1) is a compile-time constant, so the validity
    // masks fold away (step 0 does one compare and zero adj_bits gathers).
    walk_step<7, true >(0, v, window, adj_nodes32, adj_offset32, degrees32,
                        node_id32, adj_bits, choices, xrow, nrow, erow);
    walk_step<6, false>(1, v, window, adj_nodes32, adj_offset32, degrees32,
                        node_id32, adj_bits, choices, xrow, nrow, erow);
    walk_step<5, false>(2, v, window, adj_nodes32, adj_offset32, degrees32,
                        node_id32, adj_bits, choices, xrow, nrow, erow);
    walk_step<4, false>(3, v, window, adj_nodes32, adj_offset32, degrees32,
                        node_id32, adj_bits, choices, xrow, nrow, erow);
    walk_step<3, false>(4, v, window, adj_nodes32, adj_offset32, degrees32,
                        node_id32, adj_bits, choices, xrow, nrow, erow);
    walk_step<2, false>(5, v, window, adj_nodes32, adj_offset32, degrees32,
                        node_id32, adj_bits, choices, xrow, nrow, erow);
    walk_step<1, false>(6, v, window, adj_nodes32, adj_offset32, degrees32,
                        node_id32, adj_bits, choices, xrow, nrow, erow);

    // Steady state: no masking, 7 unconditional clause-able adj_bits gathers.
#pragma unroll 1
    for (int i = WIN - 1; i < STEPS; ++i)
        walk_step<0, false>(i, v, window, adj_nodes32, adj_offset32, degrees32,
                            node_id32, adj_bits, choices, xrow, nrow, erow);
}

extern "C" void kernel_launch(void* const* d_in, const int* in_sizes, int n_in,
                              void* d_out, int out_size, void* d_ws, size_t ws_size,
                              hipStream_t stream) {
    // setup_inputs() order: x, edge_index, adj_offset, degrees, node_id,
    //                       adj_bits, choices   (x is unused by the reference)
    const long long* edge_index = (const long long*)d_in[1];
    const long long* adj_nodes  = edge_index + (size_t)N_NODES * DEG; // row 1
    const unsigned*  adj_nodes32  = (const unsigned*)adj_nodes;
    const unsigned*  adj_offset32 = (const unsigned*)d_in[2];
    const unsigned*  degrees32    = (const unsigned*)d_in[3];
    const unsigned*  node_id32    = (const unsigned*)d_in[4];
    const long long* adj_bits     = (const long long*)d_in[5];
    const long long* choices      = (const long long*)d_in[6];

    float* out       = (float*)d_out;
    float* out_nodes = out;                                   // N*65
    float* out_edges = out + (size_t)N_NODES * 65;            // N*64
    float* out_x     = out + (size_t)N_NODES * (65 + 64);     // N*15*65

    const int block = 256;                                    // 8 wave32s
    const int grid  = (N_NODES + block - 1) / block;
    hipLaunchKernelGGL(Walker_12601434046503_kernel, dim3(grid), dim3(block),
                       0, stream,
                       adj_nodes32, adj_offset32, degrees32, node_id32,
                       adj_bits, choices, out_nodes, out_edges, out_x);
}